// TokenRep_5995774345607
// MI455X (gfx1250) — compile-verified
//
#include <hip/hip_runtime.h>
#include <stdint.h>

typedef float v4f __attribute__((ext_vector_type(4)));
typedef int v4i __attribute__((ext_vector_type(4)));

constexpr int B_ = 16;
constexpr int S_ = 4096;
constexpr int H_ = 768;
constexpr int T_ = 2000;
constexpr int WAVES = 8;               // 256 threads / wave32
constexpr int CHUNKS = H_ / (32 * 4);  // 6 float4 chunks per row per lane
constexpr int SCAN_ACTIVE = T_ / 8;    // 250 threads own 8 lengths each

// ---------------------------------------------------------------------------
// Kernel 1: per-batch exclusive prefix sum of subtoken_lengths -> starts
// starts[b][t] = 1 + sum(len[b][0..t-1])   (== 1 + ends - len)
// ---------------------------------------------------------------------------
__global__ __launch_bounds__(256) void tok_scan(const int* __restrict__ lens,
                                                int* __restrict__ starts) {
  __shared__ int part[256];
  const int b = blockIdx.x;
  const int tid = threadIdx.x;
  const bool active = tid < SCAN_ACTIVE;

  int v[8];
  int s = 0;
  if (active) {  // unconditional vectorized 32B load per active thread
    const v4i* L4 = (const v4i*)(lens + (size_t)b * T_) + tid * 2;
    const v4i a = L4[0];
    const v4i c = L4[1];
    v[0] = a.x; v[1] = a.y; v[2] = a.z; v[3] = a.w;
    v[4] = c.x; v[5] = c.y; v[6] = c.z; v[7] = c.w;
#pragma unroll
    for (int i = 0; i < 8; ++i) s += v[i];
  }
  part[tid] = s;  // inactive threads contribute 0
  __syncthreads();

  // Hillis-Steele inclusive scan over 256 per-thread partials
  for (int off = 1; off < 256; off <<= 1) {
    const int x = part[tid];
    const int y = (tid >= off) ? part[tid - off] : 0;
    __syncthreads();
    part[tid] = x + y;
    __syncthreads();
  }

  if (active) {
    int run = (tid > 0) ? part[tid - 1] : 0;  // exclusive batch prefix
    int o[8];
#pragma unroll
    for (int i = 0; i < 8; ++i) {
      o[i] = 1 + run;
      run += v[i];
    }
    v4i* S4 = (v4i*)(starts + (size_t)b * T_) + tid * 2;
    S4[0] = v4i{o[0], o[1], o[2], o[3]};
    S4[1] = v4i{o[4], o[5], o[6], o[7]};
  }
}

// ---------------------------------------------------------------------------
// Kernel 2: one wave32 per token. Gather len (0..2) rows of H floats via
// CDNA5 async global->LDS copies (non-temporal: zero reuse), mean them,
// nontemporal-store the output row.
// ---------------------------------------------------------------------------
__global__ __launch_bounds__(256) void tok_pool(const float* __restrict__ hidden,
                                                const int* __restrict__ lens,
                                                const int* __restrict__ starts,
                                                float* __restrict__ out) {
  __shared__ float smem[WAVES * 2 * H_];  // 2 staged rows per wave (48 KB)

  const int lane = threadIdx.x & 31;
  const int w = threadIdx.x >> 5;
  const int t = blockIdx.x * WAVES + w;  // token id within batch, [0, T)
  const int b = blockIdx.y;
  const size_t bt = (size_t)b * T_ + t;

  const int len = __builtin_amdgcn_readfirstlane(lens[bt]);
  const int start = __builtin_amdgcn_readfirstlane(starts[bt]);

  float* wbuf = smem + (size_t)w * 2 * H_;
  const uint32_t lbase = (uint32_t)(uintptr_t)wbuf;  // low 32 bits = LDS addr
  const float* row0 = hidden + ((size_t)b * S_ + (size_t)start) * H_;

  // Issue async row copies (global -> LDS), ASYNCcnt-tracked, NT-hinted.
  // Each lane moves 16B per instruction; 6 instructions = one 3072B row.
  if (len >= 1) {  // wave-uniform branch
#pragma unroll
    for (int i = 0; i < CHUNKS; ++i) {
      const uint32_t laddr = lbase + (uint32_t)((i * 32 + lane) * 16);
      const v4f* g = (const v4f*)row0 + i * 32 + lane;
      asm volatile("global_load_async_to_lds_b128 %0, %1, off th:TH_LOAD_NT"
                   :
                   : "v"(laddr), "v"(g)
                   : "memory");
    }
  }
  if (len == 2) {  // wave-uniform branch
#pragma unroll
    for (int i = 0; i < CHUNKS; ++i) {
      const uint32_t laddr =
          lbase + (uint32_t)((H_ / 4 + i * 32 + lane) * 16);
      const v4f* g = (const v4f*)(row0 + H_) + i * 32 + lane;
      asm volatile("global_load_async_to_lds_b128 %0, %1, off th:TH_LOAD_NT"
                   :
                   : "v"(laddr), "v"(g)
                   : "memory");
    }
  }
  asm volatile("s_wait_asynccnt 0" ::: "memory");

  v4f acc[CHUNKS];
#pragma unroll
  for (int i = 0; i < CHUNKS; ++i) acc[i] = v4f{0.0f, 0.0f, 0.0f, 0.0f};

  if (len >= 1) {
#pragma unroll
    for (int i = 0; i < CHUNKS; ++i)
      acc[i] += *(const v4f*)(wbuf + (i * 32 + lane) * 4);
  }
  if (len == 2) {
#pragma unroll
    for (int i = 0; i < CHUNKS; ++i)
      acc[i] += *(const v4f*)(wbuf + H_ + (i * 32 + lane) * 4);
  }

  // len==0 -> acc stays 0 and we store zeros (matches reference's where()).
  const float sc = (len == 2) ? 0.5f : 1.0f;
  v4f* dst = (v4f*)(out + bt * H_);
#pragma unroll
  for (int i = 0; i < CHUNKS; ++i) {
    v4f r = acc[i];
    r *= sc;
    __builtin_nontemporal_store(r, dst + i * 32 + lane);  // streaming output
  }
}

// ---------------------------------------------------------------------------
extern "C" void kernel_launch(void* const* d_in, const int* in_sizes, int n_in,
                              void* d_out, int out_size, void* d_ws,
                              size_t ws_size, hipStream_t stream) {
  const float* hidden = (const float*)d_in[0];  // (B,S,H) float32
  const int* lens = (const int*)d_in[1];        // (B,T) int32
  float* out = (float*)d_out;                   // (B,T,H) float32
  int* starts = (int*)d_ws;                     // B*T int32 = 128 KB scratch

  tok_scan<<<B_, 256, 0, stream>>>(lens, starts);

  dim3 grid(T_ / WAVES, B_);  // 250 x 16 blocks, 8 tokens (waves) per block
  tok_pool<<<grid, 256, 0, stream>>>(hidden, lens, starts, out);
}